// PhysicallyImplementableRandomizedOscillatorsNetwork_80582176407995
// MI455X (gfx1250) — compile-verified
//
#include <hip/hip_runtime.h>
#include <hip/hip_bf16.h>

typedef float v2f __attribute__((ext_vector_type(2)));
typedef float v8f __attribute__((ext_vector_type(8)));

#define N_INP 64
#define N_HID 512
#define BATCH 128
#define SEQL  1024
#define DT    0.042f
#define HP    516   // padded LDS row stride (floats): conflict-free b64 lane pattern

__device__ __forceinline__ v8f wmma_f32(v2f a, v2f b, v8f c) {
    return __builtin_amdgcn_wmma_f32_16x16x4_f32(false, a, false, b, (short)0, c, false, false);
}

// ---------------------------------------------------------------------------
// Kernel 0: h2hT[n, k] = h2h[k, n]  (1 MB, one-time, tiled LDS transpose)
// ---------------------------------------------------------------------------
__global__ __launch_bounds__(256) void piron_transpose_kernel(const float* __restrict__ src,
                                                              float* __restrict__ dst) {
    __shared__ float tile[32][33];
    const int lane = threadIdx.x & 31;
    const int row4 = (threadIdx.x >> 5) * 4;     // 8 groups * 4 rows = 32
    const int bx = blockIdx.x & 15;              // col tile of src
    const int by = blockIdx.x >> 4;              // row tile of src
#pragma unroll
    for (int i = 0; i < 4; ++i)
        tile[row4 + i][lane] = src[(by * 32 + row4 + i) * N_HID + bx * 32 + lane];
    __syncthreads();
#pragma unroll
    for (int i = 0; i < 4; ++i)
        dst[(bx * 32 + row4 + i) * N_HID + by * 32 + lane] = tile[lane][row4 + i];
}

// ---------------------------------------------------------------------------
// Kernel 1: i2h = tanh(x @ x2h) -> states region of d_out
// ---------------------------------------------------------------------------
__global__ __launch_bounds__(256) void piron_i2h_kernel(const float* __restrict__ x,
                                                        const float* __restrict__ x2h,
                                                        float* __restrict__ states) {
    const int lane = threadIdx.x & 31;
    const int wave = threadIdx.x >> 5;           // 0..7
    const int lo = lane & 15, hi = lane >> 4;
    const long mb = (long)blockIdx.x * 16;

    v2f a[16];
    const float* xrow = x + (mb + lo) * N_INP + 2 * hi;
#pragma unroll
    for (int s = 0; s < 16; ++s) a[s] = *(const v2f*)(xrow + 4 * s);

    for (int ntl = 0; ntl < 4; ++ntl) {
        const int col = (wave * 4 + ntl) * 16 + lo;
        v8f accA = {}, accB = {};
#pragma unroll
        for (int s = 0; s < 16; s += 2) {
            const int k0 = 4 * s + 2 * hi;
            const int k1 = 4 * (s + 1) + 2 * hi;
            v2f b0, b1;
            b0.x = x2h[k0 * N_HID + col];
            b0.y = x2h[(k0 + 1) * N_HID + col];
            b1.x = x2h[k1 * N_HID + col];
            b1.y = x2h[(k1 + 1) * N_HID + col];
            accA = wmma_f32(a[s], b0, accA);
            accB = wmma_f32(a[s + 1], b1, accB);
        }
        const v8f acc = accA + accB;
#pragma unroll
        for (int r = 0; r < 8; ++r) {
            const int m = r + 8 * hi;
            states[(mb + m) * N_HID + col] = tanhf(acc[r]);
        }
    }
}

// ---------------------------------------------------------------------------
// Kernel 2: sequential oscillator scan. 8 blocks x 16 batch rows, 16 waves.
// USE_T: phase-1 B operands come from the h2hT copy (contiguous b64 loads).
// ---------------------------------------------------------------------------
template <bool USE_T>
__global__ __launch_bounds__(512) void piron_scan_kernel(const float* __restrict__ h2h,
                                                         const float* __restrict__ h2hT,
                                                         const float* __restrict__ bias,
                                                         const float* __restrict__ gamma,
                                                         const float* __restrict__ eps,
                                                         float* __restrict__ out) {
    __shared__ __align__(16) float hy_lds[16 * HP];
    __shared__ __align__(16) float rec_lds[16 * HP];

    const int tid = threadIdx.x;
    const int lane = tid & 31;
    const int wave = tid >> 5;                   // 0..15
    const int lo = lane & 15, hi = lane >> 4;
    const int b0 = blockIdx.x * 16;

    const int col0 = wave * 32 + lo;
    const int col1 = wave * 32 + 16 + lo;
    const float bs0 = bias[col0],  bs1 = bias[col1];
    const float g0  = gamma[col0], g1  = gamma[col1];
    const float e0  = eps[col0],   e1  = eps[col1];

    for (int i = tid; i < 16 * HP; i += blockDim.x) hy_lds[i] = 0.0f;
    v8f hz0 = {}, hz1 = {};
    __syncthreads();

    for (int t = 0; t < SEQL; ++t) {
        // ---- Phase 1: rec = tanh(hy @ h2h + bias), two partial K-chains ----
        v8f a0A = {}, a0B = {}, a1A = {}, a1B = {};
#pragma unroll 2
        for (int kk = 0; kk < N_HID; kk += 8) {
            const v2f avA = *(const v2f*)&hy_lds[lo * HP + kk + 2 * hi];
            const v2f avB = *(const v2f*)&hy_lds[lo * HP + kk + 4 + 2 * hi];
            v2f b0A, b1A, b0B, b1B;
            if (USE_T) {
                b0A = *(const v2f*)&h2hT[col0 * N_HID + kk + 2 * hi];
                b1A = *(const v2f*)&h2hT[col1 * N_HID + kk + 2 * hi];
                b0B = *(const v2f*)&h2hT[col0 * N_HID + kk + 4 + 2 * hi];
                b1B = *(const v2f*)&h2hT[col1 * N_HID + kk + 4 + 2 * hi];
            } else {
                const int kA = kk + 2 * hi, kB = kk + 4 + 2 * hi;
                b0A.x = h2h[kA * N_HID + col0]; b0A.y = h2h[(kA + 1) * N_HID + col0];
                b1A.x = h2h[kA * N_HID + col1]; b1A.y = h2h[(kA + 1) * N_HID + col1];
                b0B.x = h2h[kB * N_HID + col0]; b0B.y = h2h[(kB + 1) * N_HID + col0];
                b1B.x = h2h[kB * N_HID + col1]; b1B.y = h2h[(kB + 1) * N_HID + col1];
            }
            a0A = wmma_f32(avA, b0A, a0A);
            a1A = wmma_f32(avA, b1A, a1A);
            a0B = wmma_f32(avB, b0B, a0B);
            a1B = wmma_f32(avB, b1B, a1B);
        }
        const v8f acc0 = a0A + a0B;
        const v8f acc1 = a1A + a1B;
#pragma unroll
        for (int r = 0; r < 8; ++r) {
            const int m = r + 8 * hi;
            rec_lds[m * HP + col0] = tanhf(acc0[r] + bs0);
            rec_lds[m * HP + col1] = tanhf(acc1[r] + bs1);
        }
        __syncthreads();

        // ---- Phase 2: acc = rec @ h2h^T (contiguous b64 from row-major h2h) ----
        v8f r0A = {}, r0B = {}, r1A = {}, r1B = {};
#pragma unroll 2
        for (int kk = 0; kk < N_HID; kk += 8) {
            const v2f avA = *(const v2f*)&rec_lds[lo * HP + kk + 2 * hi];
            const v2f avB = *(const v2f*)&rec_lds[lo * HP + kk + 4 + 2 * hi];
            const v2f b0A = *(const v2f*)&h2h[col0 * N_HID + kk + 2 * hi];
            const v2f b1A = *(const v2f*)&h2h[col1 * N_HID + kk + 2 * hi];
            const v2f b0B = *(const v2f*)&h2h[col0 * N_HID + kk + 4 + 2 * hi];
            const v2f b1B = *(const v2f*)&h2h[col1 * N_HID + kk + 4 + 2 * hi];
            r0A = wmma_f32(avA, b0A, r0A);
            r1A = wmma_f32(avA, b1A, r1A);
            r0B = wmma_f32(avB, b0B, r0B);
            r1B = wmma_f32(avB, b1B, r1B);
        }
        const v8f rr0 = r0A + r0B;
        const v8f rr1 = r1A + r1B;
#pragma unroll
        for (int r = 0; r < 8; ++r) {
            const int m = r + 8 * hi;
            const long row = ((long)(b0 + m) * SEQL + t) * N_HID;
            const float i0 = out[row + col0];    // i2h from kernel 1
            const float i1 = out[row + col1];
            float hy0 = hy_lds[m * HP + col0];
            float hy1 = hy_lds[m * HP + col1];
            float z0 = hz0[r], z1 = hz1[r];
            z0 += DT * (i0 - rr0[r] - g0 * hy0 - e0 * z0);
            z1 += DT * (i1 - rr1[r] - g1 * hy1 - e1 * z1);
            hz0[r] = z0; hz1[r] = z1;
            hy0 += DT * z0;
            hy1 += DT * z1;
            hy_lds[m * HP + col0] = hy0;
            hy_lds[m * HP + col1] = hy1;
            out[row + col0] = hy0;
            out[row + col1] = hy1;
        }
        __syncthreads();
    }

    // ---- hy_f epilogue ----
    float* hyf = out + (long)BATCH * SEQL * N_HID;
    for (int i = tid; i < 16 * N_HID; i += blockDim.x) {
        const int m = i >> 9, c = i & (N_HID - 1);
        hyf[(b0 + m) * N_HID + c] = hy_lds[m * HP + c];
    }
}

extern "C" void kernel_launch(void* const* d_in, const int* in_sizes, int n_in,
                              void* d_out, int out_size, void* d_ws, size_t ws_size,
                              hipStream_t stream) {
    const float* x     = (const float*)d_in[0];   // [128,1024,64]
    const float* x2h   = (const float*)d_in[1];   // [64,512]
    const float* h2h   = (const float*)d_in[2];   // [512,512]
    const float* bias  = (const float*)d_in[3];   // [512]
    const float* gamma = (const float*)d_in[4];   // [512]
    const float* eps   = (const float*)d_in[5];   // [512]
    float* out = (float*)d_out;                   // states [128,1024,512] ++ hy_f [128,512]

    piron_i2h_kernel<<<(BATCH * SEQL) / 16, 256, 0, stream>>>(x, x2h, out);

    const bool useT = ws_size >= (size_t)N_HID * N_HID * sizeof(float);
    if (useT) {
        float* h2hT = (float*)d_ws;
        piron_transpose_kernel<<<256, 256, 0, stream>>>(h2h, h2hT);
        piron_scan_kernel<true><<<BATCH / 16, 512, 0, stream>>>(h2h, h2hT, bias, gamma, eps, out);
    } else {
        piron_scan_kernel<false><<<BATCH / 16, 512, 0, stream>>>(h2h, h2h, bias, gamma, eps, out);
    }
}